// CrossAttention_68350109549162
// MI455X (gfx1250) — compile-verified
//
#include <hip/hip_runtime.h>
#include <math.h>

typedef __attribute__((ext_vector_type(2))) float v2f;
typedef __attribute__((ext_vector_type(4))) float v4f;
typedef __attribute__((ext_vector_type(8))) float v8f;

#define HDIM 256

// -----------------------------------------------------------------------------
// Tiny GEMM + bias using V_WMMA_F32_16X16X4_F32.
//   out(32,N) = in(32,K) * W(N,K)^T + bias(N)
// One wave (32 threads) per 16x16 output tile; grid = (N/16, M/16=2).
//
// fp32 WMMA layouts (CDNA5 ISA 7.12.2):
//   A 16x4 : lane l<16 -> row M=l,    VGPR0=K0, VGPR1=K1
//            lane l>=16 -> row M=l-16, VGPR0=K2, VGPR1=K3
//   B 4x16 : lane l<16 -> col N=l,    VGPR0=K0, VGPR1=K1
//            lane l>=16 -> col N=l-16, VGPR0=K2, VGPR1=K3
//   C/D    : VGPR r, lane l -> M = r + 8*(l/16), N = l%16
// -----------------------------------------------------------------------------
__global__ __launch_bounds__(32) void gemm_bias_wmma(
    const float* __restrict__ in,   // (32, K) row-major
    const float* __restrict__ W,    // (N, K) row-major
    const float* __restrict__ bias, // (N)
    float* __restrict__ out,        // (32, N) row-major
    int N, int K)
{
  const int lane = threadIdx.x;        // 0..31
  const int half = lane >> 4;          // 0 or 1 (K sub-block)
  const int l16  = lane & 15;
  const int n0 = blockIdx.x * 16;
  const int m0 = blockIdx.y * 16;

  const float* aRow = in + (size_t)(m0 + l16) * K + 2 * half;
  const float* bRow = W  + (size_t)(n0 + l16) * K + 2 * half;

  v8f c = {0.f, 0.f, 0.f, 0.f, 0.f, 0.f, 0.f, 0.f};
  for (int k = 0; k < K; k += 4) {
    v2f a = *(const v2f*)(aRow + k);   // K = 2*half, 2*half+1
    v2f b = *(const v2f*)(bRow + k);
    c = __builtin_amdgcn_wmma_f32_16x16x4_f32(
        /*neg_a=*/false, a, /*neg_b=*/false, b,
        /*c_mod=*/(short)0, c, /*reuse_a=*/false, /*reuse_b=*/false);
  }

  const float bval = bias[n0 + l16];
  float* o = out + (size_t)(n0 + l16);
#pragma unroll
  for (int r = 0; r < 8; ++r) {
    const int m = m0 + r + 8 * half;
    o[(size_t)m * N] = c[r] + bval;
  }
}

// -----------------------------------------------------------------------------
// Fused broadcast-add + LayerNorm, one wave32 per row of 768 floats.
// 24 floats/lane held in registers (6 x float4), single-pass sum/sumsq,
// wave shuffle reduction. audio / out streamed with non-temporal hints.
// -----------------------------------------------------------------------------
__global__ __launch_bounds__(256) void add_layernorm_kernel(
    const float* __restrict__ audio,  // (rows, 768)
    const float* __restrict__ proj,   // (32, 768)
    const float* __restrict__ gamma,  // (768)
    const float* __restrict__ beta,   // (768)
    float* __restrict__ out,          // (rows, 768)
    int rows)
{
  const int lane   = threadIdx.x & 31;
  const int gwave  = (blockIdx.x * blockDim.x + threadIdx.x) >> 5;
  const int nwaves = (gridDim.x * blockDim.x) >> 5;

  // gamma/beta are per-column constants for this wave: preload once.
  v4f g[6], bt[6];
#pragma unroll
  for (int i = 0; i < 6; ++i) {
    g[i]  = ((const v4f*)gamma)[lane + 32 * i];
    bt[i] = ((const v4f*)beta )[lane + 32 * i];
  }

  for (int row = gwave; row < rows; row += nwaves) {
    const int b = row >> 11;  // S = 2048
    const v4f* arow = (const v4f*)(audio + (size_t)row * 768);
    const v4f* prow = (const v4f*)(proj  + (size_t)b   * 768);

    v4f x[6];
    float sum = 0.f, sq = 0.f;
#pragma unroll
    for (int i = 0; i < 6; ++i) {
      v4f a = __builtin_nontemporal_load(&arow[lane + 32 * i]);  // streamed once
      v4f p = prow[lane + 32 * i];                               // L2-hot
      v4f xv = a + p;
      x[i] = xv;
      sum += xv.x + xv.y + xv.z + xv.w;
      sq  += xv.x * xv.x + xv.y * xv.y + xv.z * xv.z + xv.w * xv.w;
    }

#pragma unroll
    for (int off = 16; off > 0; off >>= 1) {
      sum += __shfl_xor(sum, off, 32);
      sq  += __shfl_xor(sq,  off, 32);
    }

    const float mean = sum * (1.0f / 768.0f);
    float var = sq * (1.0f / 768.0f) - mean * mean;
    var = var < 0.f ? 0.f : var;
    const float rs = 1.0f / sqrtf(var + 1e-5f);

    v4f* orow = (v4f*)(out + (size_t)row * 768);
#pragma unroll
    for (int i = 0; i < 6; ++i) {
      v4f o = (x[i] - mean) * rs * g[i] + bt[i];
      __builtin_nontemporal_store(o, &orow[lane + 32 * i]);
    }
  }
}

// -----------------------------------------------------------------------------
// Host-side launcher.
// Input order: 0 audio(32,2048,768) 1 vision(32,512) 2 Wq 3 bq 4 Wk 5 bk
//              6 Wv(256,512) 7 bv 8 in_proj_w(768,256) 9 in_proj_b(768)
//              10 Wo_mha(256,256) 11 bo_mha 12 Wout(768,256) 13 bout
//              14 gamma(768) 15 beta(768)
// (k / Wq / Wk path is dead code in the reference.)
// -----------------------------------------------------------------------------
extern "C" void kernel_launch(void* const* d_in, const int* in_sizes, int n_in,
                              void* d_out, int out_size, void* d_ws, size_t ws_size,
                              hipStream_t stream) {
  (void)in_sizes; (void)n_in; (void)out_size; (void)ws_size;

  const float* audio     = (const float*)d_in[0];
  const float* vision    = (const float*)d_in[1];
  const float* Wv        = (const float*)d_in[6];
  const float* bv        = (const float*)d_in[7];
  const float* in_proj_w = (const float*)d_in[8];
  const float* in_proj_b = (const float*)d_in[9];
  const float* Wo        = (const float*)d_in[10];
  const float* bo        = (const float*)d_in[11];
  const float* Wout      = (const float*)d_in[12];
  const float* bout      = (const float*)d_in[13];
  const float* gamma     = (const float*)d_in[14];
  const float* beta      = (const float*)d_in[15];
  float* out = (float*)d_out;

  float* ws   = (float*)d_ws;
  float* v    = ws;                  // 32 x 256
  float* v2   = ws + 32 * HDIM;      // 32 x 256
  float* attn = ws + 2 * 32 * HDIM;  // 32 x 256
  float* proj = ws + 3 * 32 * HDIM;  // 32 x 768

  // v = vision @ Wv^T + bv           (K = 512)
  gemm_bias_wmma<<<dim3(HDIM / 16, 2), 32, 0, stream>>>(vision, Wv, bv, v, HDIM, 512);
  // v2 = v @ Wiv^T + biv             (Wiv = in_proj_w[2H:], biv = in_proj_b[2H:])
  gemm_bias_wmma<<<dim3(HDIM / 16, 2), 32, 0, stream>>>(
      v, in_proj_w + 2 * HDIM * HDIM, in_proj_b + 2 * HDIM, v2, HDIM, HDIM);
  // attn = v2 @ Wo^T + bo
  gemm_bias_wmma<<<dim3(HDIM / 16, 2), 32, 0, stream>>>(v2, Wo, bo, attn, HDIM, HDIM);
  // proj = attn @ Wout^T + bout      (N = 768)
  gemm_bias_wmma<<<dim3(768 / 16, 2), 32, 0, stream>>>(attn, Wout, bout, proj, 768, HDIM);

  // out = LayerNorm(audio + proj[:,None,:]) * gamma + beta
  const int rows = 32 * 2048;
  add_layernorm_kernel<<<2048, 256, 0, stream>>>(audio, proj, gamma, beta, out, rows);
}